// NTMHead_81819126989104
// MI455X (gfx1250) — compile-verified
//
#include <hip/hip_runtime.h>
#include <math.h>

#define N_SLOTS   524288
#define DIM_MEM   128
#define DIM_IN    1024
#define NFACT     138        // 128 + 1 + 1 + 7 + 1
#define SHIFT_R   3
#define EPS_K     1e-8f
#define NBLK      1024
#define TPB       256

typedef __attribute__((ext_vector_type(2))) float v2f;
typedef __attribute__((ext_vector_type(8))) float v8f;

// ---- workspace layout (float offsets) ----
#define WS_FACTORS 0                       // 138
#define WS_PARAMS  160                     // 32
#define WS_PMAX    192                     // 1024 partial maxes
#define WS_PSUM    1216                    // 1024 partial sums (exp)
#define WS_PPOW    2240                    // 1024 partial sums (pow)
#define WS_RPART   3264                    // 1024*128 read partials
#define WS_S       (3264 + NBLK*DIM_MEM)   // N scores -> exp values
#define WS_W       (WS_S + N_SLOTS)        // N sharpened weights (unnormalized)
// total ~ (WS_W + N_SLOTS)*4 ≈ 4.8 MB

// params: [0]=clamped key norm [1]=key_s [2]=gate [3..9]=shift_w [10]=sharp
//         [11]=global max [12]=1/sum(exp) [13]=1/sum(pw)

// ---------------- factors = x @ W_wf + b_wf ----------------
__global__ void k_factors(const float* __restrict__ x, const float* __restrict__ Wf,
                          const float* __restrict__ bf, float* __restrict__ factors) {
    __shared__ float red[TPB];
    int j = blockIdx.x;
    float s = 0.f;
    for (int i = threadIdx.x; i < DIM_IN; i += TPB)
        s = fmaf(x[i], Wf[(size_t)i * NFACT + j], s);
    red[threadIdx.x] = s;
    __syncthreads();
    for (int off = TPB / 2; off > 0; off >>= 1) {
        if (threadIdx.x < off) red[threadIdx.x] += red[threadIdx.x + off];
        __syncthreads();
    }
    if (threadIdx.x == 0) factors[j] = red[0] + bf[j];
}

// ---------------- scalar params (key norm, shift softmax, sharpening) ----------------
__global__ void k_params(const float* __restrict__ f, float* __restrict__ p) {
    __shared__ float red[DIM_MEM];
    int t = threadIdx.x;                     // 128 threads
    float v = f[t];
    red[t] = v * v;
    __syncthreads();
    for (int off = 64; off > 0; off >>= 1) {
        if (t < off) red[t] += red[t + off];
        __syncthreads();
    }
    if (t == 0) {
        p[0] = fmaxf(sqrtf(red[0]), EPS_K);      // clamped key norm
        p[1] = f[DIM_MEM];                       // key strength
        p[2] = 1.0f;                             // softmax of single logit == 1
        float m = -3.4e38f;
        for (int j = 0; j < 7; ++j) m = fmaxf(m, f[DIM_MEM + 2 + j]);
        float e[7], s = 0.f;
        for (int j = 0; j < 7; ++j) { e[j] = expf(f[DIM_MEM + 2 + j] - m); s += e[j]; }
        for (int j = 0; j < 7; ++j) p[3 + j] = e[j] / s;
        p[10] = fmaxf(f[NFACT - 1], 0.f) + 1.0f; // sharpening
    }
}

// ---------------- content scores via V_WMMA_F32_16X16X4_F32 ----------------
// One wave per 16-row tile: A = 16x4 f32 row chunk, B = key chunk broadcast
// across all 16 columns, C accumulates the 16 dot products. Same A registers
// feed the per-row squared-norm accumulation -> memory streamed exactly once.
__global__ void k_scores(const float* __restrict__ mem, const float* __restrict__ factors,
                         const float* __restrict__ p, float* __restrict__ S,
                         float* __restrict__ pmax) {
    __shared__ float keyLds[DIM_MEM];
    __shared__ float wred[8];
    int tid = threadIdx.x;
    if (tid < DIM_MEM) keyLds[tid] = factors[tid];
    __syncthreads();

    const float keyn_c = p[0];
    const float key_s  = p[1];

    int lane = tid & 31;
    int wv   = tid >> 5;
    int half = lane >> 4;            // 0: K=0..1, 1: K=2..3 (A layout)
    int lrow = lane & 15;
    int gw   = blockIdx.x * 8 + wv;
    int nw   = NBLK * 8;             // 8192 waves; 32768 tiles -> exactly 4 each
    const int ntiles = N_SLOTS / 16;

    float lmax = -3.4e38f;

    for (int tile = gw; tile < ntiles; tile += nw) {
        int r0 = tile * 16;
        const float* rowp = mem + (size_t)(r0 + lrow) * DIM_MEM + 2 * half;
        if (tile + nw < ntiles)
            __builtin_prefetch(mem + (size_t)(tile + nw) * 16 * DIM_MEM + lane * 16, 0, 0);

        v8f c0 = {0.f,0.f,0.f,0.f,0.f,0.f,0.f,0.f};
        v8f c1 = {0.f,0.f,0.f,0.f,0.f,0.f,0.f,0.f};
        float nrm = 0.f;
        #pragma unroll 8
        for (int k = 0; k < DIM_MEM; k += 8) {
            v2f a0; a0.x = rowp[k];     a0.y = rowp[k + 1];
            v2f a1; a1.x = rowp[k + 4]; a1.y = rowp[k + 5];
            v2f b0; b0.x = keyLds[k + 2 * half];     b0.y = keyLds[k + 2 * half + 1];
            v2f b1; b1.x = keyLds[k + 4 + 2 * half]; b1.y = keyLds[k + 4 + 2 * half + 1];
            c0 = __builtin_amdgcn_wmma_f32_16x16x4_f32(false, a0, false, b0,
                                                       (short)0, c0, false, false);
            c1 = __builtin_amdgcn_wmma_f32_16x16x4_f32(false, a1, false, b1,
                                                       (short)0, c1, false, false);
            nrm = fmaf(a0.x, a0.x, nrm); nrm = fmaf(a0.y, a0.y, nrm);
            nrm = fmaf(a1.x, a1.x, nrm); nrm = fmaf(a1.y, a1.y, nrm);
        }
        // full ||row||^2: combine the two K-halves of each row
        float nfull = nrm + __shfl_xor(nrm, 16, 32);

        // lanes<16 hold dots of rows 0..7 in c[0..7]; lanes>=16 rows 8..15
        float sc[8];
        #pragma unroll
        for (int r = 0; r < 8; ++r) {
            float ns = __shfl(nfull, half * 8 + r, 32);   // normsq of the row this lane scores
            float denom = fmaxf(sqrtf(ns), EPS_K) * keyn_c;
            sc[r] = key_s * ((c0[r] + c1[r]) / denom);
            lmax = fmaxf(lmax, sc[r]);
        }
        if (lrow == 0) {
            float4* op = (float4*)(S + r0 + half * 8);
            op[0] = make_float4(sc[0], sc[1], sc[2], sc[3]);
            op[1] = make_float4(sc[4], sc[5], sc[6], sc[7]);
        }
    }
    // block max reduction
    for (int m = 16; m > 0; m >>= 1) lmax = fmaxf(lmax, __shfl_xor(lmax, m, 32));
    if (lane == 0) wred[wv] = lmax;
    __syncthreads();
    if (tid == 0) {
        float m = wred[0];
        for (int w = 1; w < 8; ++w) m = fmaxf(m, wred[w]);
        pmax[blockIdx.x] = m;
    }
}

// ---------------- generic deterministic reduce (mode 0: max, 1: 1/sum) ----------------
__global__ void k_reduce(const float* __restrict__ src, int n, float* __restrict__ dst, int mode) {
    __shared__ float red[TPB];
    int t = threadIdx.x;
    float a = (mode == 0) ? -3.4e38f : 0.f;
    for (int i = t; i < n; i += TPB)
        a = (mode == 0) ? fmaxf(a, src[i]) : a + src[i];
    red[t] = a;
    __syncthreads();
    for (int off = TPB / 2; off > 0; off >>= 1) {
        if (t < off) red[t] = (mode == 0) ? fmaxf(red[t], red[t + off]) : red[t] + red[t + off];
        __syncthreads();
    }
    if (t == 0) *dst = (mode == 0) ? red[0] : (1.0f / red[0]);
}

// ---------------- exp(score - max), partial sums ----------------
__global__ void k_exp(float* __restrict__ S, const float* __restrict__ p,
                      float* __restrict__ psum) {
    __shared__ float red[TPB];
    const float gmax = p[11];
    float s = 0.f;
    for (int i = blockIdx.x * TPB + threadIdx.x; i < N_SLOTS; i += NBLK * TPB) {
        float e = expf(S[i] - gmax);
        S[i] = e;
        s += e;
    }
    red[threadIdx.x] = s;
    __syncthreads();
    for (int off = TPB / 2; off > 0; off >>= 1) {
        if (threadIdx.x < off) red[threadIdx.x] += red[threadIdx.x + off];
        __syncthreads();
    }
    if (threadIdx.x == 0) psum[blockIdx.x] = red[0];
}

// ---------------- gate + circular shift convolution + sharpening ----------------
__global__ void k_shiftpow(const float* __restrict__ S, const float* __restrict__ prev,
                           const float* __restrict__ p, float* __restrict__ W,
                           float* __restrict__ ppow) {
    __shared__ float red[TPB];
    const float inv_sum = p[12], gate = p[2], sharp = p[10];
    float sw[7];
    #pragma unroll
    for (int j = 0; j < 7; ++j) sw[j] = p[3 + j];

    float acc = 0.f;
    for (int i = blockIdx.x * TPB + threadIdx.x; i < N_SLOTS; i += NBLK * TPB) {
        float sh = 0.f;
        #pragma unroll
        for (int j = 0; j < 7; ++j) {
            int t = i - (SHIFT_R - j);                // roll(gated, R-j)[i] = gated[i-(R-j)]
            t += (t < 0) ? N_SLOTS : 0;
            t -= (t >= N_SLOTS) ? N_SLOTS : 0;
            float gated = fmaf(gate, S[t] * inv_sum, (1.f - gate) * prev[t]);
            sh = fmaf(sw[j], gated, sh);
        }
        float pw = powf(sh, sharp);
        W[i] = pw;
        acc += pw;
    }
    red[threadIdx.x] = acc;
    __syncthreads();
    for (int off = TPB / 2; off > 0; off >>= 1) {
        if (threadIdx.x < off) red[threadIdx.x] += red[threadIdx.x + off];
        __syncthreads();
    }
    if (threadIdx.x == 0) ppow[blockIdx.x] = red[0];
}

// ---------------- read_vector partials: one row per wave, float4 streaming ----------------
__global__ void k_read(const float* __restrict__ mem, const float* __restrict__ Wp,
                       float* __restrict__ rpart) {
    __shared__ float acc[8][DIM_MEM];
    int tid = threadIdx.x, lane = tid & 31, wv = tid >> 5;
    int gw = blockIdx.x * 8 + wv;
    int nw = NBLK * 8;
    float4 a = make_float4(0.f, 0.f, 0.f, 0.f);
    for (int row = gw; row < N_SLOTS; row += nw) {
        if (row + nw < N_SLOTS)
            __builtin_prefetch((const float4*)(mem + (size_t)(row + nw) * DIM_MEM) + lane, 0, 0);
        float wgt = Wp[row];                               // uniform -> scalarized
        float4 m = ((const float4*)(mem + (size_t)row * DIM_MEM))[lane];
        a.x = fmaf(wgt, m.x, a.x); a.y = fmaf(wgt, m.y, a.y);
        a.z = fmaf(wgt, m.z, a.z); a.w = fmaf(wgt, m.w, a.w);
    }
    acc[wv][lane * 4 + 0] = a.x; acc[wv][lane * 4 + 1] = a.y;
    acc[wv][lane * 4 + 2] = a.z; acc[wv][lane * 4 + 3] = a.w;
    __syncthreads();
    if (tid < DIM_MEM) {
        float s = 0.f;
        for (int w = 0; w < 8; ++w) s += acc[w][tid];
        rpart[(size_t)blockIdx.x * DIM_MEM + tid] = s;
    }
}

// ---------------- final: sum block partials, apply 1/sum(pw) ----------------
__global__ void k_final(const float* __restrict__ rpart, const float* __restrict__ p,
                        float* __restrict__ out) {
    int d = threadIdx.x;                   // 128 threads
    float s = 0.f;
    for (int b = 0; b < NBLK; ++b) s += rpart[(size_t)b * DIM_MEM + d];
    out[d] = s * p[13];
}

extern "C" void kernel_launch(void* const* d_in, const int* in_sizes, int n_in,
                              void* d_out, int out_size, void* d_ws, size_t ws_size,
                              hipStream_t stream) {
    const float* x    = (const float*)d_in[0];   // (1024,)
    const float* prev = (const float*)d_in[1];   // (524288,)
    const float* mem  = (const float*)d_in[2];   // (524288,128)
    const float* Wf   = (const float*)d_in[3];   // (1024,138)
    const float* bf   = (const float*)d_in[4];   // (138,)
    float* out = (float*)d_out;                  // (128,)
    float* ws  = (float*)d_ws;

    float* factors = ws + WS_FACTORS;
    float* params  = ws + WS_PARAMS;
    float* pmax    = ws + WS_PMAX;
    float* psum    = ws + WS_PSUM;
    float* ppow    = ws + WS_PPOW;
    float* rpart   = ws + WS_RPART;
    float* S       = ws + WS_S;
    float* Wp      = ws + WS_W;

    k_factors <<<NFACT, TPB, 0, stream>>>(x, Wf, bf, factors);
    k_params  <<<1, DIM_MEM, 0, stream>>>(factors, params);
    k_scores  <<<NBLK, TPB, 0, stream>>>(mem, factors, params, S, pmax);
    k_reduce  <<<1, TPB, 0, stream>>>(pmax, NBLK, params + 11, 0);
    k_exp     <<<NBLK, TPB, 0, stream>>>(S, params, psum);
    k_reduce  <<<1, TPB, 0, stream>>>(psum, NBLK, params + 12, 1);
    k_shiftpow<<<NBLK, TPB, 0, stream>>>(S, prev, params, Wp, ppow);
    k_reduce  <<<1, TPB, 0, stream>>>(ppow, NBLK, params + 13, 1);
    k_read    <<<NBLK, TPB, 0, stream>>>(mem, Wp, rpart);
    k_final   <<<1, DIM_MEM, 0, stream>>>(rpart, params, out);
}